// TrueHopfieldLayer_21698174779710
// MI455X (gfx1250) — compile-verified
//
#include <hip/hip_runtime.h>
#include <hip/hip_bf16.h>
#include <math.h>

// ---------------------------------------------------------------------------
// TrueHopfieldLayer on MI455X (gfx1250)
// fp32 end-to-end via V_WMMA_F32_16X16X4_F32, fed by CDNA5 async global->LDS
// (global_load_async_to_lds_b128 / s_wait_asynccnt) with double buffering.
// ---------------------------------------------------------------------------

typedef __attribute__((ext_vector_type(2))) float v2f;
typedef __attribute__((ext_vector_type(8))) float v8f;

#define HOP_B      4
#define HOP_S      2048
#define HOP_H      512
#define HOP_SCALE  0.044194173824159216f   /* 1/sqrt(512) */
#define HOP_ITERS  5
#define HOP_THRESH 1e-4f

#define KC      32          // K-chunk staged in LDS
#define LDA_PAD 36          // row pitch (floats) for [row][k] tiles: 144B = 9*16B
#define LDB_PAD 68          // row pitch (floats) for [k][n] tiles:   272B = 17*16B

// Async copy of 16 bytes global -> LDS (ASYNCcnt-tracked, GVS addressing).
__device__ __forceinline__ void async_b128(unsigned lds_byte_off,
                                           const float* __restrict__ sbase,
                                           unsigned gbyte_off)
{
    asm volatile("global_load_async_to_lds_b128 %0, %1, %2 offset:0"
                 :
                 : "v"(lds_byte_off), "v"(gbyte_off), "s"(sbase)
                 : "memory");
}
#define ASYNC_WAIT_0() asm volatile("s_wait_asynccnt 0x0" ::: "memory")
#define ASYNC_WAIT_8() asm volatile("s_wait_asynccnt 0x8" ::: "memory")

__device__ __forceinline__ unsigned lds_off_of(const void* p)
{
    return (unsigned)(unsigned long long)p;   // low 32 bits = LDS byte offset
}

// ---------------------------------------------------------------------------
// GEMM (NT): C[M,N] = alpha * A[M,K] @ B[N,K]^T + bias[N]
// Block = 128 threads = 4 waves; each wave computes a 16x64 strip of C.
// WMMA fragment layouts (16x16x4 f32):
//   A 16x4 : lane = M + 16*(K/2), vgpr = K%2  -> contiguous float2 per lane
//   B 4x16 : lane = N + 16*(K/2), vgpr = K%2  -> NT: contiguous float2
//   C 16x16: vgpr r, lanes 0-15 -> (M=r, N=lane); lanes 16-31 -> (M=r+8)
// ---------------------------------------------------------------------------
__global__ __launch_bounds__(128)
void gemm_nt_wmma(const float* __restrict__ A, const float* __restrict__ Bm,
                  const float* __restrict__ bias, float* __restrict__ C,
                  int M, int N, int K, float alpha)
{
    __shared__ float LA[2][64 * LDA_PAD];
    __shared__ float LB[2][64 * LDA_PAD];

    const int tid  = threadIdx.x;
    const int lane = tid & 31;
    const int wave = tid >> 5;

    const int m0 = blockIdx.y * 64 + wave * 16;
    const int n0 = blockIdx.x * 64;

    const int cq    = lane & 15;
    const int khalf = (lane >> 4) << 1;           // 0 or 2
    const int arow  = wave * 16 + cq;             // tile-local A row for frags

    // Per-thread async-load assignment: half a 128B row per thread.
    const int lr = tid >> 1;                      // tile row 0..63
    const int lh = (tid & 1) * 16;                // half-row offset in floats

    const unsigned aTile = (unsigned)(blockIdx.y * 64 + lr) * K;  // elem index
    const unsigned bTile = (unsigned)(n0 + lr) * K;

    v8f acc[4] = {v8f{}, v8f{}, v8f{}, v8f{}};

    const int nchunks = K / KC;

    // ---- prologue: stage chunk 0 into buffer 0 ----
    {
        unsigned lA = lds_off_of(&LA[0][lr * LDA_PAD + lh]);
        unsigned lB = lds_off_of(&LB[0][lr * LDA_PAD + lh]);
        unsigned ga = (aTile + lh) * 4u;
        unsigned gb = (bTile + lh) * 4u;
#pragma unroll
        for (int i = 0; i < 4; ++i) {
            async_b128(lA + 16u * i, A,  ga + 16u * i);
            async_b128(lB + 16u * i, Bm, gb + 16u * i);
        }
    }

    for (int c = 0; c < nchunks; ++c) {
        const int cur = c & 1;
        if (c + 1 < nchunks) {
            const int nxt = (c + 1) & 1;
            const unsigned k0 = (unsigned)(c + 1) * KC;
            unsigned lA = lds_off_of(&LA[nxt][lr * LDA_PAD + lh]);
            unsigned lB = lds_off_of(&LB[nxt][lr * LDA_PAD + lh]);
            unsigned ga = (aTile + k0 + lh) * 4u;
            unsigned gb = (bTile + k0 + lh) * 4u;
#pragma unroll
            for (int i = 0; i < 4; ++i) {
                async_b128(lA + 16u * i, A,  ga + 16u * i);
                async_b128(lB + 16u * i, Bm, gb + 16u * i);
            }
            ASYNC_WAIT_8();      // chunk c landed; only chunk c+1 outstanding
        } else {
            ASYNC_WAIT_0();
        }
        __syncthreads();

#pragma unroll
        for (int ks = 0; ks < KC / 4; ++ks) {
            const int kk = ks * 4 + khalf;
            v2f a = *(const v2f*)(&LA[cur][arow * LDA_PAD + kk]);
#pragma unroll
            for (int j = 0; j < 4; ++j) {
                v2f bfr = *(const v2f*)(&LB[cur][(j * 16 + cq) * LDA_PAD + kk]);
                acc[j] = __builtin_amdgcn_wmma_f32_16x16x4_f32(
                    false, a, false, bfr, (short)0, acc[j], false, false);
            }
        }
        __syncthreads();         // buffer 'cur' free for chunk c+2
    }

    const int rbase = m0 + ((lane >> 4) << 3);    // +8 rows for upper half-wave
#pragma unroll
    for (int j = 0; j < 4; ++j) {
        const int col = n0 + 16 * j + cq;
        const float bb = bias ? bias[col] : 0.0f;
#pragma unroll
        for (int r = 0; r < 8; ++r) {
            C[(long)(rbase + r) * N + col] = alpha * acc[j][r] + bb;
        }
    }
}

// ---------------------------------------------------------------------------
// GEMM (NN): C[M,N] = A[M,K] @ B[K,N]   (B row-major, ldb = N)
// B chunk staged in memory order [k][n] with 16B-aligned padded pitch.
// ---------------------------------------------------------------------------
__global__ __launch_bounds__(128)
void gemm_nn_wmma(const float* __restrict__ A, const float* __restrict__ Bm,
                  float* __restrict__ C, int M, int N, int K)
{
    __shared__ float LA[2][64 * LDA_PAD];
    __shared__ float LB[2][KC * LDB_PAD];

    const int tid  = threadIdx.x;
    const int lane = tid & 31;
    const int wave = tid >> 5;

    const int m0 = blockIdx.y * 64 + wave * 16;
    const int n0 = blockIdx.x * 64;

    const int cq    = lane & 15;
    const int khalf = (lane >> 4) << 1;
    const int arow  = wave * 16 + cq;

    // A staging: half a row per thread (as in NT).
    const int lr = tid >> 1;
    const int lh = (tid & 1) * 16;
    const unsigned aTile = (unsigned)(blockIdx.y * 64 + lr) * K;

    // B staging: quarter of a 64-float row per thread; 32 K-rows per chunk.
    const int br = tid >> 2;                      // k-row 0..31
    const int bq = (tid & 3) * 16;                // offset in floats

    v8f acc[4] = {v8f{}, v8f{}, v8f{}, v8f{}};

    const int nchunks = K / KC;

    {
        unsigned lA = lds_off_of(&LA[0][lr * LDA_PAD + lh]);
        unsigned lB = lds_off_of(&LB[0][br * LDB_PAD + bq]);
        unsigned ga = (aTile + lh) * 4u;
        unsigned gb = ((unsigned)br * N + n0 + bq) * 4u;
#pragma unroll
        for (int i = 0; i < 4; ++i) {
            async_b128(lA + 16u * i, A,  ga + 16u * i);
            async_b128(lB + 16u * i, Bm, gb + 16u * i);
        }
    }

    for (int c = 0; c < nchunks; ++c) {
        const int cur = c & 1;
        if (c + 1 < nchunks) {
            const int nxt = (c + 1) & 1;
            const unsigned k0 = (unsigned)(c + 1) * KC;
            unsigned lA = lds_off_of(&LA[nxt][lr * LDA_PAD + lh]);
            unsigned lB = lds_off_of(&LB[nxt][br * LDB_PAD + bq]);
            unsigned ga = (aTile + k0 + lh) * 4u;
            unsigned gb = ((k0 + (unsigned)br) * N + n0 + bq) * 4u;
#pragma unroll
            for (int i = 0; i < 4; ++i) {
                async_b128(lA + 16u * i, A,  ga + 16u * i);
                async_b128(lB + 16u * i, Bm, gb + 16u * i);
            }
            ASYNC_WAIT_8();
        } else {
            ASYNC_WAIT_0();
        }
        __syncthreads();

#pragma unroll
        for (int ks = 0; ks < KC / 4; ++ks) {
            const int kk = ks * 4 + khalf;
            v2f a = *(const v2f*)(&LA[cur][arow * LDA_PAD + kk]);
#pragma unroll
            for (int j = 0; j < 4; ++j) {
                v2f bfr;
                bfr.x = LB[cur][kk * LDB_PAD + j * 16 + cq];
                bfr.y = LB[cur][(kk + 1) * LDB_PAD + j * 16 + cq];
                acc[j] = __builtin_amdgcn_wmma_f32_16x16x4_f32(
                    false, a, false, bfr, (short)0, acc[j], false, false);
            }
        }
        __syncthreads();
    }

    const int rbase = m0 + ((lane >> 4) << 3);
#pragma unroll
    for (int j = 0; j < 4; ++j) {
        const int col = n0 + 16 * j + cq;
#pragma unroll
        for (int r = 0; r < 8; ++r) {
            C[(long)(rbase + r) * N + col] = acc[j][r];
        }
    }
}

// ---------------------------------------------------------------------------
// Row softmax (numerically stable), one block of 256 threads (8 waves) per row.
// ---------------------------------------------------------------------------
__global__ __launch_bounds__(256)
void softmax_rows(float* __restrict__ S, int ncols)
{
    __shared__ float red[8];
    float* row = S + (long)blockIdx.x * ncols;
    const int tid = threadIdx.x;

    float m = -INFINITY;
    for (int i = tid; i < ncols; i += 256) m = fmaxf(m, row[i]);
#pragma unroll
    for (int off = 16; off; off >>= 1) m = fmaxf(m, __shfl_xor(m, off, 32));
    if ((tid & 31) == 0) red[tid >> 5] = m;
    __syncthreads();
    if (tid == 0) {
        float v = red[0];
#pragma unroll
        for (int i = 1; i < 8; ++i) v = fmaxf(v, red[i]);
        red[0] = v;
    }
    __syncthreads();
    m = red[0];
    __syncthreads();

    float s = 0.0f;
    for (int i = tid; i < ncols; i += 256) {
        float e = __expf(row[i] - m);
        row[i] = e;
        s += e;
    }
#pragma unroll
    for (int off = 16; off; off >>= 1) s += __shfl_xor(s, off, 32);
    if ((tid & 31) == 0) red[tid >> 5] = s;
    __syncthreads();
    if (tid == 0) {
        float v = 0.0f;
#pragma unroll
        for (int i = 0; i < 8; ++i) v += red[i];
        red[0] = v;
    }
    __syncthreads();
    const float inv = 1.0f / red[0];
    for (int i = tid; i < ncols; i += 256) row[i] *= inv;
}

// ---------------------------------------------------------------------------
// Frobenius-norm accumulators: sc[0] += ||new-old||^2 ; sc[1] += ||old||^2
// ---------------------------------------------------------------------------
__global__ __launch_bounds__(256)
void norm_reduce(const float* __restrict__ nw, const float* __restrict__ od,
                 float* __restrict__ sc, long n)
{
    long i0 = (long)blockIdx.x * blockDim.x + threadIdx.x;
    long stride = (long)gridDim.x * blockDim.x;
    float d2 = 0.0f, o2 = 0.0f;
    for (long i = i0; i < n; i += stride) {
        float d = nw[i] - od[i];
        float o = od[i];
        d2 += d * d;
        o2 += o * o;
    }
#pragma unroll
    for (int off = 16; off; off >>= 1) {
        d2 += __shfl_xor(d2, off, 32);
        o2 += __shfl_xor(o2, off, 32);
    }
    if ((threadIdx.x & 31) == 0) {
        atomicAdd(&sc[0], d2);
        atomicAdd(&sc[1], o2);
    }
}

__global__ void init_flags(float* sc, int* flags)
{
    sc[0] = 0.0f;
    sc[1] = 0.0f;
    flags[0] = 0;   // converged
    flags[1] = 0;   // iterations
}

// Copy new_state -> state only while not yet converged (freeze semantics).
__global__ __launch_bounds__(256)
void commit_state(float* __restrict__ state, const float* __restrict__ nst,
                  const int* __restrict__ flags, long n)
{
    if (flags[0]) return;
    long i0 = (long)blockIdx.x * blockDim.x + threadIdx.x;
    long stride = (long)gridDim.x * blockDim.x;
    for (long i = i0; i < n; i += stride) state[i] = nst[i];
}

// Convergence bookkeeping (matches the jax scan step exactly).
__global__ void gate_kernel(float* sc, int* flags)
{
    if (!flags[0]) {
        flags[1] += 1;
        float delta = sqrtf(sc[0]) / (sqrtf(sc[1]) + 1e-8f);
        if (delta < HOP_THRESH) flags[0] = 1;
    }
    sc[0] = 0.0f;   // reset accumulators for the next iteration
    sc[1] = 0.0f;
}

__global__ void write_tail(float* out, const int* flags, long off)
{
    out[off]     = (float)flags[1];
    out[off + 1] = flags[0] ? 1.0f : 0.0f;
}

// ---------------------------------------------------------------------------
// Host-side orchestration
// ---------------------------------------------------------------------------
extern "C" void kernel_launch(void* const* d_in, const int* in_sizes, int n_in,
                              void* d_out, int out_size, void* d_ws, size_t ws_size,
                              hipStream_t stream)
{
    const float* x  = (const float*)d_in[0];
    const float* Wq = (const float*)d_in[1];
    const float* bq = (const float*)d_in[2];
    const float* Wk = (const float*)d_in[3];
    const float* bk = (const float*)d_in[4];
    const float* Wv = (const float*)d_in[5];
    const float* bv = (const float*)d_in[6];
    const float* Wo = (const float*)d_in[7];
    const float* bo = (const float*)d_in[8];

    const long SH  = (long)HOP_S * HOP_H;          // per-batch state elems
    const long BSH = (long)HOP_B * SH;             // 4,194,304
    const long SS  = (long)HOP_S * HOP_S;          // 4,194,304

    float* ws     = (float*)d_ws;
    float* state  = ws;                            // [B,S,H]
    float* keys   = ws + 1 * BSH;                  // [B,S,H]
    float* vals   = ws + 2 * BSH;                  // [B,S,H]
    float* nstate = ws + 3 * BSH;                  // [B,S,H]
    float* outt   = ws + 4 * BSH;                  // [B,S,H]
    float* scores = ws + 5 * BSH;                  // [S,S]  (per-batch reuse)
    float* sc     = ws + 5 * BSH + SS;             // 2 norm accumulators
    int*   flags  = (int*)(sc + 2);                // converged, iterations

    const dim3 blk(128);
    const dim3 gproj(HOP_H / 64, (HOP_B * HOP_S) / 64, 1);  // 8 x 128
    const dim3 gscore(HOP_S / 64, HOP_S / 64, 1);           // 32 x 32
    const dim3 gupd(HOP_H / 64, HOP_S / 64, 1);             // 8 x 32

    // ---- Projections: query/keys/values ----
    gemm_nt_wmma<<<gproj, blk, 0, stream>>>(x, Wq, bq, state,
                                            HOP_B * HOP_S, HOP_H, HOP_H, 1.0f);
    gemm_nt_wmma<<<gproj, blk, 0, stream>>>(x, Wk, bk, keys,
                                            HOP_B * HOP_S, HOP_H, HOP_H, 1.0f);
    gemm_nt_wmma<<<gproj, blk, 0, stream>>>(x, Wv, bv, vals,
                                            HOP_B * HOP_S, HOP_H, HOP_H, 1.0f);
    init_flags<<<1, 1, 0, stream>>>(sc, flags);

    // ---- Hopfield iterations (fixed-length, device-side freeze) ----
    for (int it = 0; it < HOP_ITERS; ++it) {
        for (int b = 0; b < HOP_B; ++b) {
            gemm_nt_wmma<<<gscore, blk, 0, stream>>>(state + b * SH, keys + b * SH,
                                                     nullptr, scores,
                                                     HOP_S, HOP_S, HOP_H, HOP_SCALE);
            softmax_rows<<<HOP_S, 256, 0, stream>>>(scores, HOP_S);
            gemm_nn_wmma<<<gupd, blk, 0, stream>>>(scores, keys + b * SH,
                                                   nstate + b * SH,
                                                   HOP_S, HOP_H, HOP_S);
        }
        norm_reduce<<<1024, 256, 0, stream>>>(nstate, state, sc, BSH);
        commit_state<<<1024, 256, 0, stream>>>(state, nstate, flags, BSH);
        gate_kernel<<<1, 1, 0, stream>>>(sc, flags);
    }

    // ---- Final readout attention against values ----
    for (int b = 0; b < HOP_B; ++b) {
        gemm_nt_wmma<<<gscore, blk, 0, stream>>>(state + b * SH, keys + b * SH,
                                                 nullptr, scores,
                                                 HOP_S, HOP_S, HOP_H, HOP_SCALE);
        softmax_rows<<<HOP_S, 256, 0, stream>>>(scores, HOP_S);
        gemm_nn_wmma<<<gupd, blk, 0, stream>>>(scores, vals + b * SH,
                                               outt + b * SH,
                                               HOP_S, HOP_H, HOP_S);
    }

    // ---- Output projection straight into d_out ----
    gemm_nt_wmma<<<gproj, blk, 0, stream>>>(outt, Wo, bo, (float*)d_out,
                                            HOP_B * HOP_S, HOP_H, HOP_H, 1.0f);

    // ---- Tuple tail: (iterations, converged) ----
    if ((long)out_size >= BSH + 2)
        write_tail<<<1, 1, 0, stream>>>((float*)d_out, flags, BSH);
}